// GATBlock_19980187861404
// MI455X (gfx1250) — compile-verified
//
#include <hip/hip_runtime.h>
#include <hip/hip_bf16.h>
#include <stdint.h>

typedef __attribute__((ext_vector_type(2))) float v2f;
typedef __attribute__((ext_vector_type(8))) float v8f;

// ---------- helpers: order-preserving float<->uint for atomic max ----------
__device__ __forceinline__ unsigned enc_f32(float f) {
    unsigned u = __float_as_uint(f);
    return (u & 0x80000000u) ? ~u : (u | 0x80000000u);
}
__device__ __forceinline__ float dec_f32(unsigned u) {
    unsigned b = (u & 0x80000000u) ? (u & 0x7FFFFFFFu) : ~u;
    return __uint_as_float(b);
}
#define ENC_NEG_INF 0x007FFFFFu  // enc(-inf)

// =====================================================================
// Kernel 1: h = x@W  (to ws), out_base = x@skip_W + skip_b + conv_bias (to d_out)
// One block = 16 rows of x. 8 waves: wave w -> gemm = w>>2, col tile = (w&3)*16.
// fp32 WMMA 16x16x4, 16 k-steps over K=64.
// =====================================================================
__global__ void __launch_bounds__(256)
gat_node_gemm(const float* __restrict__ x, const float* __restrict__ W,
              const float* __restrict__ skip_W, const float* __restrict__ conv_bias,
              const float* __restrict__ skip_b, float* __restrict__ h,
              float* __restrict__ out, int n)
{
    const int wave = threadIdx.x >> 5;
    const int lane = threadIdx.x & 31;
    const int m0   = blockIdx.x * 16;
    const int gemm = wave >> 2;              // 0: W -> h, 1: skip_W -> out base
    const int n0   = (wave & 3) * 16;        // head tile (C == 16)
    const float* __restrict__ B = gemm ? skip_W : W;

    const int lrow = lane & 15;
    const int kq   = (lane >> 4) * 2;        // lanes 0-15: K+{0,1}; 16-31: K+{2,3}
    int arow = m0 + lrow; if (arow >= n) arow = n - 1;   // clamp: keep EXEC full
    const int col  = n0 + lrow;

    v8f c = {0.f,0.f,0.f,0.f,0.f,0.f,0.f,0.f};
    const float* __restrict__ ap = x + (size_t)arow * 64 + kq;

    #pragma unroll
    for (int kk = 0; kk < 16; ++kk) {
        const int kb = kk * 4;
        v2f a, b;
        a.x = ap[kb + 0];
        a.y = ap[kb + 1];
        b.x = B[(size_t)(kb + kq + 0) * 64 + col];
        b.y = B[(size_t)(kb + kq + 1) * 64 + col];
        // D = A(16x4,f32) * B(4x16,f32) + C  ->  v_wmma_f32_16x16x4_f32
        c = __builtin_amdgcn_wmma_f32_16x16x4_f32(false, a, false, b,
                                                  (short)0, c, false, false);
    }

    const int rbase = m0 + ((lane >> 4) << 3);   // lanes 16-31 hold M = r+8
    if (gemm == 0) {
        #pragma unroll
        for (int r = 0; r < 8; ++r) {
            const int row = rbase + r;
            if (row < n) h[(size_t)row * 64 + col] = c[r];
        }
    } else {
        const float bias = conv_bias[col] + skip_b[col];
        #pragma unroll
        for (int r = 0; r < 8; ++r) {
            const int row = rbase + r;
            if (row < n) out[(size_t)row * 64 + col] = c[r] + bias;
        }
    }
}

// =====================================================================
// Kernel 2: per (node, head) attention coefficients + state init
// =====================================================================
__global__ void gat_coeff(const float* __restrict__ h, const float* __restrict__ att_src,
                          const float* __restrict__ att_dst, float* __restrict__ a_src,
                          float* __restrict__ a_dst, unsigned* __restrict__ emax_u,
                          float* __restrict__ denom, int n)
{
    const int t = blockIdx.x * blockDim.x + threadIdx.x;
    if (t >= n * 4) return;
    const int node = t >> 2, hd = t & 3;
    const float4* __restrict__ hp = (const float4*)(h + (size_t)node * 64 + hd * 16);
    const float4* __restrict__ as = (const float4*)(att_src + hd * 16);
    const float4* __restrict__ ad = (const float4*)(att_dst + hd * 16);
    float s = 0.f, d = 0.f;
    #pragma unroll
    for (int i = 0; i < 4; ++i) {
        float4 hv = hp[i], a1 = as[i], a2 = ad[i];
        s += hv.x*a1.x + hv.y*a1.y + hv.z*a1.z + hv.w*a1.w;
        d += hv.x*a2.x + hv.y*a2.y + hv.z*a2.z + hv.w*a2.w;
    }
    a_src[t] = s;
    a_dst[t] = d;
    emax_u[t] = ENC_NEG_INF;
    denom[t]  = 0.f;
}

__device__ __forceinline__ void edge_sd(const long long* __restrict__ ei,
                                        int t, int E, int& s, int& d)
{
    if (t < E) { s = (int)ei[t]; d = (int)ei[(size_t)E + t]; }
    else       { s = d = t - E; }               // self loops appended
}

// =====================================================================
// Kernel 3: segment max of leaky-relu logits via ordered-uint atomicMax
// =====================================================================
__global__ void gat_edge_max(const long long* __restrict__ ei,
                             const float* __restrict__ a_src, const float* __restrict__ a_dst,
                             unsigned* __restrict__ emax_u, int E, int n)
{
    const int t = blockIdx.x * blockDim.x + threadIdx.x;
    if (t >= E + n) return;
    int s, d; edge_sd(ei, t, E, s, d);
    const float4 as = *(const float4*)(a_src + (size_t)s * 4);
    const float4 ad = *(const float4*)(a_dst + (size_t)d * 4);
    unsigned* __restrict__ em = emax_u + (size_t)d * 4;
    #pragma unroll
    for (int hd = 0; hd < 4; ++hd) {
        float v = (&as.x)[hd] + (&ad.x)[hd];
        v = v > 0.f ? v : 0.2f * v;
        atomicMax(&em[hd], enc_f32(v));
    }
}

// =====================================================================
// Kernel 4: e_exp = exp(e - max[dst]); denom += e_exp; stash e_exp per edge
// =====================================================================
__global__ void gat_edge_exp(const long long* __restrict__ ei,
                             const float* __restrict__ a_src, const float* __restrict__ a_dst,
                             const unsigned* __restrict__ emax_u, float* __restrict__ denom,
                             float4* __restrict__ aexp, int E, int n)
{
    const int t = blockIdx.x * blockDim.x + threadIdx.x;
    if (t >= E + n) return;
    int s, d; edge_sd(ei, t, E, s, d);
    const float4 as = *(const float4*)(a_src + (size_t)s * 4);
    const float4 ad = *(const float4*)(a_dst + (size_t)d * 4);
    float4 ex;
    float* __restrict__ dn = denom + (size_t)d * 4;
    #pragma unroll
    for (int hd = 0; hd < 4; ++hd) {
        float v = (&as.x)[hd] + (&ad.x)[hd];
        v = v > 0.f ? v : 0.2f * v;
        float e = expf(v - dec_f32(emax_u[(size_t)d * 4 + hd]));
        (&ex.x)[hd] = e;
        atomicAdd(&dn[hd], e);
    }
    aexp[t] = ex;
}

// =====================================================================
// Kernel 5: scatter  out[dst] += h[src] * (e_exp / denom[dst])
// 16 lanes per edge (one per channel), 4 heads per lane.
// =====================================================================
__global__ void gat_scatter(const long long* __restrict__ ei, const float* __restrict__ h,
                            const float4* __restrict__ aexp, const float* __restrict__ denom,
                            float* __restrict__ out, int E, int n)
{
    const int t = blockIdx.x * blockDim.x + threadIdx.x;
    const int eid = t >> 4;
    const int ch  = t & 15;
    if (eid >= E + n) return;
    int s, d; edge_sd(ei, eid, E, s, d);
    const float4 ax = aexp[eid];
    const float4 dn = *(const float4*)(denom + (size_t)d * 4);
    const float* __restrict__ hs = h + (size_t)s * 64;
    float* __restrict__ op = out + (size_t)d * 64;
    #pragma unroll
    for (int hd = 0; hd < 4; ++hd) {
        const float alpha = (&ax.x)[hd] / (&dn.x)[hd];
        atomicAdd(&op[hd * 16 + ch], hs[hd * 16 + ch] * alpha);
    }
}

// =====================================================================
// Kernel 6: ELU(alpha=0.1) in place
// =====================================================================
__global__ void gat_elu(float* __restrict__ out, int total)
{
    const int t = blockIdx.x * blockDim.x + threadIdx.x;
    if (t >= total) return;
    const float v = out[t];
    out[t] = v > 0.f ? v : 0.1f * expm1f(fminf(v, 0.f));
}

// =====================================================================
extern "C" void kernel_launch(void* const* d_in, const int* in_sizes, int n_in,
                              void* d_out, int out_size, void* d_ws, size_t ws_size,
                              hipStream_t stream)
{
    const float*     x         = (const float*)d_in[0];
    const long long* edge_idx  = (const long long*)d_in[1];   // int64 [2,E]
    const float*     W         = (const float*)d_in[2];
    const float*     att_src   = (const float*)d_in[3];
    const float*     att_dst   = (const float*)d_in[4];
    const float*     conv_bias = (const float*)d_in[5];
    const float*     skip_W    = (const float*)d_in[6];
    const float*     skip_b    = (const float*)d_in[7];
    float*           out       = (float*)d_out;

    const int N = in_sizes[0] / 64;
    const int E = in_sizes[1] / 2;
    const int T = E + N;                      // edges incl. self loops

    // ---- workspace carving (≈29.6 MB) ----
    float*    h_ws     = (float*)d_ws;                         // N*64
    float*    asrc_ws  = h_ws    + (size_t)N * 64;             // N*4
    float*    adst_ws  = asrc_ws + (size_t)N * 4;              // N*4
    unsigned* emax_ws  = (unsigned*)(adst_ws + (size_t)N * 4); // N*4
    float*    denom_ws = (float*)(emax_ws + (size_t)N * 4);    // N*4
    float4*   aexp_ws  = (float4*)(denom_ws + (size_t)N * 4);  // T float4s

    const int BLK = 256;
    // 1: GEMMs (WMMA) — 16 rows per block, 8 waves cover 2 gemms x 4 head tiles
    gat_node_gemm<<<(N + 15) / 16, BLK, 0, stream>>>(x, W, skip_W, conv_bias,
                                                     skip_b, h_ws, out, N);
    // 2: attention coefficients + init
    gat_coeff<<<(N * 4 + BLK - 1) / BLK, BLK, 0, stream>>>(h_ws, att_src, att_dst,
                                                           asrc_ws, adst_ws,
                                                           emax_ws, denom_ws, N);
    // 3: segment max
    gat_edge_max<<<(T + BLK - 1) / BLK, BLK, 0, stream>>>(edge_idx, asrc_ws, adst_ws,
                                                          emax_ws, E, N);
    // 4: exp + denom
    gat_edge_exp<<<(T + BLK - 1) / BLK, BLK, 0, stream>>>(edge_idx, asrc_ws, adst_ws,
                                                          emax_ws, denom_ws, aexp_ws, E, N);
    // 5: weighted scatter-add
    gat_scatter<<<((long long)T * 16 + BLK - 1) / BLK, BLK, 0, stream>>>(edge_idx, h_ws,
                                                                         aexp_ws, denom_ws,
                                                                         out, E, N);
    // 6: ELU
    gat_elu<<<(N * 64 + BLK - 1) / BLK, BLK, 0, stream>>>(out, N * 64);
}